// SelfAttentionRiskPredictor_8624294330625
// MI455X (gfx1250) — compile-verified
//
#include <hip/hip_runtime.h>
#include <math.h>

// MI455X / gfx1250, wave32.
// Exact algebraic collapse of the rank-1-token self-attention:
//   risk[b] = alpha * mean_q( S1_q / S0_q ) + beta
//   S1_q = sum_k exp(t_q*x_k - m_q) * x_k ; S0_q = sum_k exp(t_q*x_k - m_q)
//   t_q = A*x_q + C ; m_q = max(t_q*xmax, t_q*xmin)
// The (S1,S0) reductions run on the matrix pipe via V_WMMA_F32_16X16X4_F32
// (B column 0 = x chunk, column 1 = ones), co-executing with v_exp_f32.
// Round-2 change: all x-chunk values are preloaded from LDS into 64 VGPRs in
// an unrolled prolog, and the B fragment is derived from the same registers,
// so the fully-unrolled hot loop (4 cndmask + 16 fma/exp + 8 wmma per chunk)
// contains no DS ops and no s_wait_dscnt stalls.

typedef float v2f __attribute__((ext_vector_type(2)));
typedef float v8f __attribute__((ext_vector_type(8)));

#define D_MODEL 32
#define SEQ 128

// ---------------- prep: fold all weights into 4 scalars ----------------
__global__ void risk_prep_kernel(const float* __restrict__ Wt, const float* __restrict__ bt,
                                 const float* __restrict__ Wq, const float* __restrict__ bq,
                                 const float* __restrict__ Wk, const float* __restrict__ bk,
                                 const float* __restrict__ Wv, const float* __restrict__ bv,
                                 const float* __restrict__ Wo, const float* __restrict__ bo,
                                 float* __restrict__ cst) {
  const int e = threadIdx.x;  // 0..31, one lane per d_model column
  float wqp = 0.f, bqp = 0.f, wkp = 0.f, bkp = 0.f, wvp = 0.f, bvp = 0.f;
  for (int d = 0; d < D_MODEL; ++d) {
    const float wt = Wt[d];
    const float bd = bt[d];
    const float q = Wq[d * D_MODEL + e];
    const float k = Wk[d * D_MODEL + e];
    const float v = Wv[d * D_MODEL + e];
    wqp += wt * q;  bqp += bd * q;
    wkp += wt * k;  bkp += bd * k;
    wvp += wt * v;  bvp += bd * v;
  }
  bqp += bq[e];
  bkp += bk[e];
  bvp += bv[e];
  const float wo = Wo[e];
  float aA  = wqp * wkp;   // -> A  = (wq'.wk')/sqrt(32)
  float aC  = bqp * wkp;   // -> C  = (bq'.wk')/sqrt(32)
  float aAl = wvp * wo;    // -> alpha = wv'.Wo
  float aBe = bvp * wo;    // -> beta  = bv'.Wo + bo
  for (int off = 16; off > 0; off >>= 1) {
    aA  += __shfl_xor(aA,  off, 32);
    aC  += __shfl_xor(aC,  off, 32);
    aAl += __shfl_xor(aAl, off, 32);
    aBe += __shfl_xor(aBe, off, 32);
  }
  if (e == 0) {
    const float inv_s = 0.17677669529663688f;  // 1/sqrt(32)
    cst[0] = aA * inv_s;
    cst[1] = aC * inv_s;
    cst[2] = aAl;
    cst[3] = aBe + bo[0];
  }
}

// ---------------- main: one wave32 per batch row ----------------
__global__ __launch_bounds__(256) void risk_attn_kernel(const float* __restrict__ x,
                                                        const float* __restrict__ cst,
                                                        float* __restrict__ out, int B) {
  __shared__ float4 sx[8 * 32];  // 8 waves * 128 floats
  const int lane = threadIdx.x & 31;
  const int wid  = threadIdx.x >> 5;
  const int b    = blockIdx.x * 8 + wid;
  if (b >= B) return;  // wave-uniform: EXEC stays all-ones for WMMA

  const float Acoef = cst[0], Ccoef = cst[1], alpha = cst[2], beta = cst[3];

  // Stage this batch's 128 x values in LDS (one float4 per lane, coalesced b128).
  const float4* xrow = reinterpret_cast<const float4*>(x + (size_t)b * SEQ);
  const float4 mine = xrow[lane];
  sx[wid * 32 + lane] = mine;

  // Wave-wide xmax/xmin for the exact softmax max-shift.
  float xmax = fmaxf(fmaxf(mine.x, mine.y), fmaxf(mine.z, mine.w));
  float xmin = fminf(fminf(mine.x, mine.y), fminf(mine.z, mine.w));
  for (int off = 16; off > 0; off >>= 1) {
    xmax = fmaxf(xmax, __shfl_xor(xmax, off, 32));
    xmin = fminf(xmin, __shfl_xor(xmin, off, 32));
  }

  const float* xs = reinterpret_cast<const float*>(&sx[wid * 32]);
  const float L2E = 1.4426950408889634f;
  const int m = lane & 15;  // row within a 16-q block
  const bool hi = lane >= 16;  // lanes 16..31 carry K=2,3 of each 4-wide chunk

  // Per-lane logit slope/shift (already in log2 domain) for its 8 q rows.
  float t2[8], m2[8];
#pragma unroll
  for (int qb = 0; qb < 8; ++qb) {
    const float xq = xs[qb * 16 + m];
    const float t  = Acoef * xq + Ccoef;
    t2[qb] = t * L2E;
    m2[qb] = fmaxf(t * xmax, t * xmin) * L2E;
  }

  // Preload the per-lane A-column inputs for all 32 chunks into VGPRs so the
  // hot loop touches no memory.  32 uniform ds_load_b128 issued back-to-back;
  // the scheduler staggers the dscnt waits across the unrolled sequence.
  float xa0[32], xa1[32];
#pragma unroll
  for (int kc = 0; kc < 32; ++kc) {
    const float4 xc = sx[wid * 32 + kc];
    xa0[kc] = hi ? xc.z : xc.x;  // A-frag VGPR0 column (K = hi?2:0)
    xa1[kc] = hi ? xc.w : xc.y;  // A-frag VGPR1 column (K = hi?3:1)
  }

  const v8f vzero = {0.f, 0.f, 0.f, 0.f, 0.f, 0.f, 0.f, 0.f};
  v8f acc[8];
#pragma unroll
  for (int qb = 0; qb < 8; ++qb) acc[qb] = vzero;

  // Loop-invariant part of the B fragment (1.0 at lanes 1 and 17, else 0).
  const float ones_col = (m == 1) ? 1.f : 0.f;

  // Hot loop: fully unrolled, register-resident, no DS/VMEM ops.
#pragma unroll
  for (int kc = 0; kc < 32; ++kc) {
    // B (4x16): col 0 = x chunk, col 1 = ones, rest zero.
    // Lane 0 needs x[4kc] == xa0[kc] (low half); lane 16 needs x[4kc+2]
    // == xa0[kc] (high half) -- same register, so B is 2 cndmasks per VGPR.
    const float b0 = (m == 0) ? xa0[kc] : ones_col;
    const float b1 = (m == 0) ? xa1[kc] : ones_col;
    const v2f Bf = {b0, b1};

#pragma unroll
    for (int qb = 0; qb < 8; ++qb) {
      const v2f Af = {__builtin_amdgcn_exp2f(t2[qb] * xa0[kc] - m2[qb]),
                      __builtin_amdgcn_exp2f(t2[qb] * xa1[kc] - m2[qb])};
      // D = A(16x4) x B(4x16) + C : col0 += E*x, col1 += E
      acc[qb] = __builtin_amdgcn_wmma_f32_16x16x4_f32(
          false, Af, false, Bf, (short)0, acc[qb], false, false);
    }
  }

  // Extract y_q = S1_q / S0_q.  D VGPR r: lane0=D[r][0], lane1=D[r][1],
  // lane16=D[r+8][0], lane17=D[r+8][1].  S0 >= 1 (max-shifted), rcp is safe.
  float part = 0.f;
#pragma unroll
  for (int qb = 0; qb < 8; ++qb) {
#pragma unroll
    for (int r = 0; r < 8; ++r) {
      const float dnum = acc[qb][r];
      const float dden = __shfl_xor(dnum, 1, 32);
      const float ratio = dnum * __builtin_amdgcn_rcpf(dden);
      part += (m == 0) ? ratio : 0.f;  // valid at lanes 0 and 16
    }
  }
  part += __shfl_xor(part, 16, 32);
  if (lane == 0) out[b] = alpha * (part * (1.0f / 128.0f)) + beta;
}

// ---------------- launcher ----------------
extern "C" void kernel_launch(void* const* d_in, const int* in_sizes, int n_in,
                              void* d_out, int out_size, void* d_ws, size_t ws_size,
                              hipStream_t stream) {
  const float* x  = (const float*)d_in[0];
  const float* Wt = (const float*)d_in[1];
  const float* bt = (const float*)d_in[2];
  const float* Wq = (const float*)d_in[3];
  const float* bq = (const float*)d_in[4];
  const float* Wk = (const float*)d_in[5];
  const float* bk = (const float*)d_in[6];
  const float* Wv = (const float*)d_in[7];
  const float* bv = (const float*)d_in[8];
  const float* Wo = (const float*)d_in[9];
  const float* bo = (const float*)d_in[10];
  float* out = (float*)d_out;
  float* cst = (float*)d_ws;  // 4 folded scalars {A, C, alpha, beta}

  const int B = in_sizes[0] / SEQ;

  hipLaunchKernelGGL(risk_prep_kernel, dim3(1), dim3(32), 0, stream,
                     Wt, bt, Wq, bq, Wk, bk, Wv, bv, Wo, bo, cst);

  const int blocks = (B + 7) / 8;  // 8 waves (8 batch rows) per block
  hipLaunchKernelGGL(risk_attn_kernel, dim3(blocks), dim3(256), 0, stream,
                     x, cst, out, B);
}